// MultiHeadAttentionBiLSTMCRFNER_4071628997374
// MI455X (gfx1250) — compile-verified
//
#include <hip/hip_runtime.h>
#include <math.h>

// ---------------------------------------------------------------------------
// MI455X / gfx1250, wave32. fp32 WMMA (V_WMMA_F32_16X16X4_F32) for all GEMMs
// (preserves fp32 reference numerics; CRF logsumexp is precision-sensitive).
// Big GEMMs stage the shared W tile in LDS via async global->LDS copies
// (GLOBAL_LOAD_ASYNC_TO_LDS_B128, ASYNCcnt) with double buffering.
// ---------------------------------------------------------------------------

typedef float v2f __attribute__((ext_vector_type(2)));
typedef float v8f __attribute__((ext_vector_type(8)));
typedef int   v4i __attribute__((ext_vector_type(4)));

#define BB 8
#define SS 512
#define DD 768
#define HH 768
#define CC 9
#define NHH 8
#define DHH 96
#define MTOK (BB * SS)   // 4096 tokens

#if defined(__has_builtin)
#if __has_builtin(__builtin_amdgcn_global_load_async_to_lds_b128) && \
    __has_builtin(__builtin_amdgcn_s_wait_asynccnt)
#define HAVE_ASYNC_LDS 1
#endif
#endif

#if defined(HAVE_ASYNC_LDS) && defined(__AMDGCN__)
typedef __attribute__((address_space(1))) v4i* gas1_v4i;
typedef __attribute__((address_space(3))) v4i* las3_v4i;
#endif

__device__ __forceinline__ v8f wmma4(v2f a, v2f b, v8f c) {
  // D = A(16x4) * B(4x16) + C(16x16), all f32
  return __builtin_amdgcn_wmma_f32_16x16x4_f32(
      /*neg_a=*/false, a, /*neg_b=*/false, b,
      /*c_mod=*/(short)0, c, /*reuse_a=*/false, /*reuse_b=*/false);
}

__device__ __forceinline__ float sigm(float x) { return 1.0f / (1.0f + expf(-x)); }

// ---------------------------------------------------------------------------
// Kernel 1: fused embedding gather + highway layer.
//   x  = emb[ids]; hh = relu(x@Wh^T+bh); t = sigmoid(x@Wt^T+bt)
//   x0 = t*hh + (1-t)*x
// grid (M/16, 768/64), block 32 (one wave); wave computes 16x64 for both GEMMs.
// ---------------------------------------------------------------------------
__global__ void k_embed_highway(const int* __restrict__ ids,
                                const float* __restrict__ emb,
                                const float* __restrict__ Wh, const float* __restrict__ bh,
                                const float* __restrict__ Wt, const float* __restrict__ bt,
                                float* __restrict__ x0) {
  const int lane = threadIdx.x & 31;
  const int khf  = lane >> 4;      // which K pair (0:{0,1} 1:{2,3})
  const int l16  = lane & 15;
  const int m0   = blockIdx.x * 16;
  const int n0b  = blockIdx.y * 64;

  const int rowid = ids[m0 + l16];               // vocab row for this A lane
  const float* arow = emb + (size_t)rowid * DD;

  v8f accH[4], accT[4];
#pragma unroll
  for (int t = 0; t < 4; ++t) {
    const float bvh = bh[n0b + t * 16 + l16];
    const float bvt = bt[n0b + t * 16 + l16];
#pragma unroll
    for (int v = 0; v < 8; ++v) { accH[t][v] = bvh; accT[t][v] = bvt; }
  }

  for (int k0 = 0; k0 < DD; k0 += 4) {
    const int kk = k0 + 2 * khf;
    v2f a; a.x = arow[kk]; a.y = arow[kk + 1];
#pragma unroll
    for (int t = 0; t < 4; ++t) {
      const int n = n0b + t * 16 + l16;
      v2f b1; b1.x = Wh[(size_t)n * DD + kk]; b1.y = Wh[(size_t)n * DD + kk + 1];
      v2f b2; b2.x = Wt[(size_t)n * DD + kk]; b2.y = Wt[(size_t)n * DD + kk + 1];
      accH[t] = wmma4(a, b1, accH[t]);
      accT[t] = wmma4(a, b2, accT[t]);
    }
  }

#pragma unroll
  for (int t = 0; t < 4; ++t) {
    const int n = n0b + t * 16 + l16;
#pragma unroll
    for (int v = 0; v < 8; ++v) {
      const int m = m0 + v + 8 * khf;
      const float xe = emb[(size_t)ids[m] * DD + n];
      const float hv = fmaxf(accH[t][v], 0.0f);
      const float tv = sigm(accT[t][v]);
      x0[(size_t)m * DD + n] = tv * hv + (1.0f - tv) * xe;
    }
  }
}

// ---------------------------------------------------------------------------
// Generic GEMM: C[M,N] = A[M,K] @ W[N,K]^T + bias[N]   (PyTorch Linear layout)
// Block = 256 threads (8 waves): block tile 128x64; all 8 waves share one
// 64-col x 32-k W tile staged in LDS (layout [col][k], stride 36 -> every 16B
// async transfer aligned, ds_load_b64 reads bank-conflict-free). Double
// buffered: async copy of chunk c+1 overlaps WMMA on chunk c.
// K must be a multiple of 32; M multiple of 128; N multiple of 64.
// ---------------------------------------------------------------------------
#define KC   32
#define BPAD 36
__global__ void k_gemm_bias(const float* __restrict__ A, const float* __restrict__ W,
                            const float* __restrict__ bias, float* __restrict__ C,
                            int M, int N, int K) {
  __shared__ float Bst[2][64 * BPAD];   // 18 KB
  const int tid  = threadIdx.x;
  const int wv   = tid >> 5;
  const int lane = tid & 31;
  const int khf  = lane >> 4;
  const int l16  = lane & 15;
  const int m0   = blockIdx.x * 128 + wv * 16;
  const int n0b  = blockIdx.y * 64;

  v8f acc[4];
#pragma unroll
  for (int t = 0; t < 4; ++t) {
    const float bv = bias[n0b + t * 16 + l16];
#pragma unroll
    for (int v = 0; v < 8; ++v) acc[t][v] = bv;
  }

  const float* arow = A + (size_t)(m0 + l16) * K;

  // stage one 64x32 W chunk into LDS buffer `buf` (512 x 16B transfers)
  auto stage = [&](int k0, int buf) {
#pragma unroll
    for (int j = tid; j < 512; j += 256) {
      const int col = j >> 3, seg = j & 7;
      const float* g = W + (size_t)(n0b + col) * K + k0 + seg * 4;
      float* l = &Bst[buf][col * BPAD + seg * 4];
#if defined(HAVE_ASYNC_LDS) && defined(__AMDGCN__)
      __builtin_amdgcn_global_load_async_to_lds_b128((gas1_v4i)g, (las3_v4i)l, 0, 0);
#else
      *(float4*)l = *(const float4*)g;
#endif
    }
  };

  stage(0, 0);
#if defined(HAVE_ASYNC_LDS) && defined(__AMDGCN__)
  __builtin_amdgcn_s_wait_asynccnt(0);
#endif
  __syncthreads();

  const int nch = K / KC;
  for (int c = 0; c < nch; ++c) {
    if (c + 1 < nch) stage((c + 1) * KC, (c + 1) & 1);
    const float* Bb = &Bst[c & 1][0];
    const int kbase = c * KC;
#pragma unroll
    for (int k0 = 0; k0 < KC; k0 += 4) {
      const int kk = k0 + 2 * khf;
      v2f a; a.x = arow[kbase + kk]; a.y = arow[kbase + kk + 1];
#pragma unroll
      for (int t = 0; t < 4; ++t) {
        const float* bcol = Bb + (t * 16 + l16) * BPAD;
        v2f b; b.x = bcol[kk]; b.y = bcol[kk + 1];
        acc[t] = wmma4(a, b, acc[t]);
      }
    }
#if defined(HAVE_ASYNC_LDS) && defined(__AMDGCN__)
    __builtin_amdgcn_s_wait_asynccnt(0);
#endif
    __syncthreads();
  }

#pragma unroll
  for (int t = 0; t < 4; ++t)
#pragma unroll
    for (int v = 0; v < 8; ++v)
      C[(size_t)(m0 + v + 8 * khf) * N + n0b + t * 16 + l16] = acc[t][v];
}

// ---------------------------------------------------------------------------
// Kernel 3: multi-head attention, one block per (b, head, 16-query block).
// qkv: [B,S,3D] (q at +0, k at +D, v at +2D within last dim; head-major DH=96).
// Scores tile P[16,512] staged in LDS (stride 516 -> conflict-free P-as-A).
// ---------------------------------------------------------------------------
#define LDSP 516
__global__ void k_attention(const float* __restrict__ qkv, float* __restrict__ o) {
  __shared__ float P[16 * LDSP];   // ~33 KB
  const int bid  = blockIdx.x;
  const int rb   = bid & 31;          // query row block (32 per b,h)
  const int h    = (bid >> 5) & 7;
  const int b    = bid >> 8;
  const int wv   = threadIdx.x >> 5;  // 8 waves
  const int lane = threadIdx.x & 31;
  const int khf  = lane >> 4;
  const int l16  = lane & 15;
  const int q0   = rb * 16;
  const float scale = 0.1020620726f;  // 1/sqrt(96)
  const size_t base = (size_t)b * SS * (3 * DD);

  // --- scores = Q @ K^T * scale; wave wv covers 64 key-columns ---
#pragma unroll 1
  for (int t = 0; t < 4; ++t) {
    const int n0 = wv * 64 + t * 16;
    v8f acc = {};
    for (int k0 = 0; k0 < DHH; k0 += 4) {
      const int kk = k0 + 2 * khf;
      v2f a, bb;
      a.x  = qkv[base + (size_t)(q0 + l16) * (3 * DD) + h * DHH + kk];
      a.y  = qkv[base + (size_t)(q0 + l16) * (3 * DD) + h * DHH + kk + 1];
      bb.x = qkv[base + (size_t)(n0 + l16) * (3 * DD) + DD + h * DHH + kk];
      bb.y = qkv[base + (size_t)(n0 + l16) * (3 * DD) + DD + h * DHH + kk + 1];
      acc = wmma4(a, bb, acc);
    }
#pragma unroll
    for (int v = 0; v < 8; ++v)
      P[(v + 8 * khf) * LDSP + n0 + l16] = acc[v] * scale;
  }
  __syncthreads();

  // --- row softmax over 512; wave wv handles rows 2wv, 2wv+1 ---
#pragma unroll
  for (int rr = 0; rr < 2; ++rr) {
    const int r = wv * 2 + rr;
    float mx = -1e30f;
    for (int j = lane; j < SS; j += 32) mx = fmaxf(mx, P[r * LDSP + j]);
#pragma unroll
    for (int off = 16; off; off >>= 1) mx = fmaxf(mx, __shfl_xor(mx, off, 32));
    float sum = 0.0f;
    for (int j = lane; j < SS; j += 32) {
      const float e = expf(P[r * LDSP + j] - mx);
      P[r * LDSP + j] = e;
      sum += e;
    }
#pragma unroll
    for (int off = 16; off; off >>= 1) sum += __shfl_xor(sum, off, 32);
    const float inv = 1.0f / sum;
    for (int j = lane; j < SS; j += 32) P[r * LDSP + j] *= inv;
  }
  __syncthreads();

  // --- O = P @ V : 6 column tiles (DH=96), waves 0..5 ---
  if (wv < 6) {
    const int n0 = wv * 16;
    v8f acc = {};
    for (int k0 = 0; k0 < SS; k0 += 4) {
      const int kk = k0 + 2 * khf;
      v2f a, bb;
      a.x  = P[l16 * LDSP + kk];
      a.y  = P[l16 * LDSP + kk + 1];
      bb.x = qkv[base + (size_t)kk * (3 * DD) + 2 * DD + h * DHH + n0 + l16];
      bb.y = qkv[base + (size_t)(kk + 1) * (3 * DD) + 2 * DD + h * DHH + n0 + l16];
      acc = wmma4(a, bb, acc);
    }
#pragma unroll
    for (int v = 0; v < 8; ++v) {
      const int row = q0 + v + 8 * khf;
      o[((size_t)b * SS + row) * DD + h * DHH + n0 + l16] = acc[v];
    }
  }
}

// ---------------------------------------------------------------------------
// Kernel 4: one BiLSTM direction per block (grid=2). Persistent workgroup of
// 768 threads = 24 waves; each wave owns 2 blocks of 16 hidden units, so all
// four gates (i,f,g,o) of its units live in its own WMMA accumulators and the
// cell state stays in registers. h state lives in LDS (padded stride 772).
// Batch (8, zero-padded to 16) is the WMMA M dimension.
// ---------------------------------------------------------------------------
#define HPAD 772
__global__ void k_lstm_dir(const float* __restrict__ in, int IN,
                           const float* __restrict__ Wih,  // [2, 3072, IN]
                           const float* __restrict__ Whh,  // [2, 3072, 768]
                           const float* __restrict__ bias, // [2, 3072]
                           float* __restrict__ out) {      // [B,S,1536]
  __shared__ float hld[8 * HPAD];
  const int dir  = blockIdx.x;
  const float* Wd = Wih + (size_t)dir * 4 * HH * IN;
  const float* Ud = Whh + (size_t)dir * 4 * HH * HH;
  const float* bd = bias + (size_t)dir * 4 * HH;
  const int wv   = threadIdx.x >> 5;   // 0..23
  const int lane = threadIdx.x & 31;
  const int khf  = lane >> 4;
  const int l16  = lane & 15;
  const int rclamp = l16 & 7;          // clamped batch row for padded lanes

  for (int i = threadIdx.x; i < 8 * HPAD; i += blockDim.x) hld[i] = 0.0f;
  float creg[2][8];
#pragma unroll
  for (int u = 0; u < 2; ++u)
#pragma unroll
    for (int v = 0; v < 8; ++v) creg[u][v] = 0.0f;
  float hnew[2][8];
  __syncthreads();

  for (int step = 0; step < SS; ++step) {
    const int t = dir ? (SS - 1 - step) : step;
#pragma unroll 1
    for (int ub = 0; ub < 2; ++ub) {
      const int u0 = (wv * 2 + ub) * 16;
      v8f acc[4];
#pragma unroll
      for (int g = 0; g < 4; ++g) {
        const float bv = bd[g * HH + u0 + l16];
#pragma unroll
        for (int v = 0; v < 8; ++v) acc[g][v] = bv;
      }
      // input contribution: x_t @ Wih^T
      const float* xrow = in + ((size_t)rclamp * SS + t) * IN;
      for (int k0 = 0; k0 < IN; k0 += 4) {
        const int kk = k0 + 2 * khf;
        v2f a;
        a.x = (l16 < 8) ? xrow[kk]     : 0.0f;
        a.y = (l16 < 8) ? xrow[kk + 1] : 0.0f;
#pragma unroll
        for (int g = 0; g < 4; ++g) {
          const float* wrow = Wd + (size_t)(g * HH + u0 + l16) * IN;
          v2f b; b.x = wrow[kk]; b.y = wrow[kk + 1];
          acc[g] = wmma4(a, b, acc[g]);
        }
      }
      // recurrent contribution: h_{t-1} @ Whh^T (h from LDS)
      for (int k0 = 0; k0 < HH; k0 += 4) {
        const int kk = k0 + 2 * khf;
        v2f a;
        a.x = (l16 < 8) ? hld[rclamp * HPAD + kk]     : 0.0f;
        a.y = (l16 < 8) ? hld[rclamp * HPAD + kk + 1] : 0.0f;
#pragma unroll
        for (int g = 0; g < 4; ++g) {
          const float* urow = Ud + (size_t)(g * HH + u0 + l16) * HH;
          v2f b; b.x = urow[kk]; b.y = urow[kk + 1];
          acc[g] = wmma4(a, b, acc[g]);
        }
      }
      // gate math: lanes 0..15 hold batch rows 0..7 in VGPRs 0..7
      if (khf == 0) {
#pragma unroll
        for (int v = 0; v < 8; ++v) {
          const float ig = sigm(acc[0][v]);
          const float fg = sigm(acc[1][v]);
          const float gg = tanhf(acc[2][v]);
          const float og = sigm(acc[3][v]);
          const float c  = fg * creg[ub][v] + ig * gg;
          creg[ub][v] = c;
          hnew[ub][v] = og * tanhf(c);
        }
      }
    }
    __syncthreads();   // all reads of hld complete
    if (khf == 0) {
#pragma unroll
      for (int ub = 0; ub < 2; ++ub) {
        const int u = (wv * 2 + ub) * 16 + l16;
#pragma unroll
        for (int v = 0; v < 8; ++v) {
          hld[v * HPAD + u] = hnew[ub][v];
          out[((size_t)v * SS + t) * (2 * HH) + dir * HH + u] = hnew[ub][v];
        }
      }
    }
    __syncthreads();   // writes visible before next step
  }
}

// ---------------------------------------------------------------------------
// Kernel 5: emissions = l1out @ fc_W^T + fc_b  ([4096,1536] x [9,1536]^T)
// ---------------------------------------------------------------------------
__global__ void k_fc(const float* __restrict__ A, const float* __restrict__ W,
                     const float* __restrict__ b, float* __restrict__ em) {
  const int idx = blockIdx.x * blockDim.x + threadIdx.x;
  if (idx >= MTOK * CC) return;
  const int m = idx / CC, c = idx % CC;
  const float* a = A + (size_t)m * (2 * HH);
  const float* w = W + (size_t)c * (2 * HH);
  float s = b[c];
  for (int k = 0; k < 2 * HH; ++k) s += a[k] * w[k];
  em[idx] = s;
}

// ---------------------------------------------------------------------------
// Kernel 6: CRF negative log-likelihood (sum over batch) -> out[0]
// ---------------------------------------------------------------------------
__global__ void k_crf_nll(const float* __restrict__ em, const int* __restrict__ lab,
                          const float* __restrict__ st, const float* __restrict__ en,
                          const float* __restrict__ tr, float* __restrict__ out) {
  __shared__ float a[BB][CC], an[BB][CC], red[BB];
  const int tid = threadIdx.x, b = tid >> 5, c = tid & 31;
  const bool act = (b < BB) && (c < CC);
  if (act) a[b][c] = st[c] + em[((size_t)b * SS) * CC + c];
  __syncthreads();
  for (int s = 1; s < SS; ++s) {
    if (act) {
      float mx = -1e30f;
#pragma unroll
      for (int p = 0; p < CC; ++p) mx = fmaxf(mx, a[b][p] + tr[p * CC + c]);
      float sm = 0.0f;
#pragma unroll
      for (int p = 0; p < CC; ++p) sm += expf(a[b][p] + tr[p * CC + c] - mx);
      an[b][c] = mx + logf(sm) + em[((size_t)b * SS + s) * CC + c];
    }
    __syncthreads();
    if (act) a[b][c] = an[b][c];
    __syncthreads();
  }
  if (act && c == 0) {
    float mx = -1e30f;
#pragma unroll
    for (int p = 0; p < CC; ++p) mx = fmaxf(mx, a[b][p] + en[p]);
    float sm = 0.0f;
#pragma unroll
    for (int p = 0; p < CC; ++p) sm += expf(a[b][p] + en[p] - mx);
    const float logZ = mx + logf(sm);
    const int* lb = lab + b * SS;
    float num = st[lb[0]] + en[lb[SS - 1]];
    for (int s = 0; s < SS; ++s) num += em[((size_t)b * SS + s) * CC + lb[s]];
    for (int s = 0; s < SS - 1; ++s) num += tr[lb[s] * CC + lb[s + 1]];
    red[b] = logZ - num;
  }
  __syncthreads();
  if (tid == 0) {
    float L = 0.0f;
    for (int i = 0; i < BB; ++i) L += red[i];
    out[0] = L;
  }
}

// ---------------------------------------------------------------------------
// Kernel 7: Viterbi decode -> out[1 + b*S + s] (tags as floats)
// ---------------------------------------------------------------------------
__global__ void k_viterbi(const float* __restrict__ em, const float* __restrict__ st,
                          const float* __restrict__ en, const float* __restrict__ tr,
                          int* __restrict__ bp, float* __restrict__ out) {
  __shared__ float a[BB][CC], an[BB][CC];
  __shared__ int lastT[BB];
  const int tid = threadIdx.x, b = tid >> 5, c = tid & 31;
  const bool act = (b < BB) && (c < CC);
  if (act) a[b][c] = st[c] + em[((size_t)b * SS) * CC + c];
  __syncthreads();
  for (int s = 1; s < SS; ++s) {
    if (act) {
      float best = -1e30f; int barg = 0;
#pragma unroll
      for (int p = 0; p < CC; ++p) {
        const float v = a[b][p] + tr[p * CC + c];
        if (v > best) { best = v; barg = p; }
      }
      an[b][c] = best + em[((size_t)b * SS + s) * CC + c];
      bp[((s - 1) * BB + b) * CC + c] = barg;
    }
    __syncthreads();
    if (act) a[b][c] = an[b][c];
    __syncthreads();
  }
  if (act && c == 0) {
    float best = -1e30f; int barg = 0;
#pragma unroll
    for (int p = 0; p < CC; ++p) {
      const float v = a[b][p] + en[p];
      if (v > best) { best = v; barg = p; }
    }
    lastT[b] = barg;
  }
  __syncthreads();
  if (tid < BB) {
    int tag = lastT[tid];
    out[1 + (size_t)tid * SS + (SS - 1)] = (float)tag;
    for (int s = SS - 2; s >= 0; --s) {
      tag = bp[(s * BB + tid) * CC + tag];
      out[1 + (size_t)tid * SS + s] = (float)tag;
    }
  }
}

// ---------------------------------------------------------------------------
// Workspace layout (floats), overlaid:
//   OFF_X   : x0 (emb+highway)   -> later: out_proj result (LSTM input)
//   OFF_QKV : qkv [4096,2304]    -> later: layer-0 BiLSTM output [4096,1536]
//   OFF_O   : attention O        -> later: layer-1 BiLSTM output [4096,1536]
//   OFF_EM  : emissions [4096,9]
//   OFF_BP  : viterbi backpointers (511*8*9 ints)
// Total ~76 MB.
// ---------------------------------------------------------------------------
static const size_t OFF_X   = 0;
static const size_t OFF_QKV = OFF_X   + (size_t)MTOK * DD;        //  3,145,728
static const size_t OFF_O   = OFF_QKV + (size_t)MTOK * 3 * DD;    // 12,582,912
static const size_t OFF_EM  = OFF_O   + (size_t)MTOK * 2 * HH;    // 18,874,368
static const size_t OFF_BP  = OFF_EM  + (size_t)MTOK * CC;        // 18,911,232

extern "C" void kernel_launch(void* const* d_in, const int* in_sizes, int n_in,
                              void* d_out, int out_size, void* d_ws, size_t ws_size,
                              hipStream_t stream) {
  const int*   ids        = (const int*)d_in[0];
  const int*   labels     = (const int*)d_in[1];
  const float* emb        = (const float*)d_in[2];
  const float* hw_Wh      = (const float*)d_in[3];
  const float* hw_bh      = (const float*)d_in[4];
  const float* hw_Wt      = (const float*)d_in[5];
  const float* hw_bt      = (const float*)d_in[6];
  const float* in_proj_w  = (const float*)d_in[7];
  const float* in_proj_b  = (const float*)d_in[8];
  const float* out_proj_w = (const float*)d_in[9];
  const float* out_proj_b = (const float*)d_in[10];
  const float* l0_Wih     = (const float*)d_in[11];
  const float* l0_Whh     = (const float*)d_in[12];
  const float* l0_b       = (const float*)d_in[13];
  const float* l1_Wih     = (const float*)d_in[14];
  const float* l1_Whh     = (const float*)d_in[15];
  const float* l1_b       = (const float*)d_in[16];
  const float* fc_W       = (const float*)d_in[17];
  const float* fc_b       = (const float*)d_in[18];
  const float* crf_start  = (const float*)d_in[19];
  const float* crf_end    = (const float*)d_in[20];
  const float* crf_trans  = (const float*)d_in[21];

  float* ws   = (float*)d_ws;
  float* x0   = ws + OFF_X;
  float* qkv  = ws + OFF_QKV;
  float* obuf = ws + OFF_O;
  float* em   = ws + OFF_EM;
  int*   bp   = (int*)(ws + OFF_BP);
  float* dout = (float*)d_out;

  // 1. embedding + highway -> x0
  k_embed_highway<<<dim3(MTOK / 16, DD / 64), 32, 0, stream>>>(
      ids, emb, hw_Wh, hw_bh, hw_Wt, hw_bt, x0);
  // 2. QKV projection -> qkv  (block tile 128x64, 8 waves, LDS-staged W)
  k_gemm_bias<<<dim3(MTOK / 128, (3 * DD) / 64), 256, 0, stream>>>(
      x0, in_proj_w, in_proj_b, qkv, MTOK, 3 * DD, DD);
  // 3. attention -> obuf  (B*NH*(S/16) = 2048 blocks)
  k_attention<<<BB * NHH * (SS / 16), 256, 0, stream>>>(qkv, obuf);
  // 4. output projection -> x0 (x0 region is dead after QKV)
  k_gemm_bias<<<dim3(MTOK / 128, DD / 64), 256, 0, stream>>>(
      obuf, out_proj_w, out_proj_b, x0, MTOK, DD, DD);
  // 5. BiLSTM layer 0: x0 -> qkv region ([B,S,1536]); grid 2 = fwd/bwd
  k_lstm_dir<<<2, 768, 0, stream>>>(x0, DD, l0_Wih, l0_Whh, l0_b, qkv);
  // 6. BiLSTM layer 1: qkv region -> obuf region ([B,S,1536])
  k_lstm_dir<<<2, 768, 0, stream>>>(qkv, 2 * HH, l1_Wih, l1_Whh, l1_b, obuf);
  // 7. FC -> emissions
  k_fc<<<(MTOK * CC + 255) / 256, 256, 0, stream>>>(obuf, fc_W, fc_b, em);
  // 8. CRF NLL -> dout[0]
  k_crf_nll<<<1, 256, 0, stream>>>(em, labels, crf_start, crf_end, crf_trans, dout);
  // 9. Viterbi decode -> dout[1..4096]
  k_viterbi<<<1, 256, 0, stream>>>(em, crf_start, crf_end, crf_trans, bp, dout);
}